// MultiHeadAttention_48275432407838
// MI455X (gfx1250) — compile-verified
//
#include <hip/hip_runtime.h>

// ---------------------------------------------------------------------------
// Multi-head attention forward for gfx1250 (MI455X), fp32 end to end using
// V_WMMA_F32_16X16X4_F32 (keeps the reference's fp32 accumulation semantics).
// GEMM tile staging uses the CDNA5 async global->LDS path (ASYNCcnt) with
// double buffering when the toolchain exposes the builtins.
// ---------------------------------------------------------------------------

typedef __attribute__((ext_vector_type(2))) float v2f;
typedef __attribute__((ext_vector_type(4))) float v4f;
typedef __attribute__((ext_vector_type(8))) float v8f;

#define D_MODEL 1024
#define N_HEADS 16
#define D_HEAD  64
#define BATCH   2
#define SEQ     2048
#define M_TOTAL (BATCH * SEQ)   // 4096 rows for the projection GEMMs

static __device__ __forceinline__ v8f wmma_f32(v2f a, v2f b, v8f c) {
    // 8 args: (neg_a, A, neg_b, B, c_mod, C, reuse_a, reuse_b)
    return __builtin_amdgcn_wmma_f32_16x16x4_f32(
        false, a, false, b, (short)0, c, false, false);
}

// ---- CDNA5 async global->LDS copy (GLOBAL_LOAD_ASYNC_TO_LDS_B128) ---------
#if __has_builtin(__builtin_amdgcn_global_load_async_to_lds_b128) && \
    __has_builtin(__builtin_amdgcn_s_wait_asynccnt)
#define HAVE_ASYNC_LDS 1
#endif

// Builtin signature (from the compiler diagnostic):
//   (v4i addrspace(1)*, v4i addrspace(3)*, imm int offset, imm int cpol)
typedef int v4i_vs __attribute__((vector_size(16)));
typedef v4i_vs __attribute__((address_space(1)))* as1_v4i_ptr;
typedef v4i_vs __attribute__((address_space(3)))* as3_v4i_ptr;

static __device__ __forceinline__ void copy16(const float* g, float* l) {
#ifdef HAVE_ASYNC_LDS
    // generic->AS1 is the identity mapping; generic LDS addresses carry the
    // LDS byte offset in the low 32 bits (aperture rules, ISA 10.2).
    __builtin_amdgcn_global_load_async_to_lds_b128(
        (as1_v4i_ptr)(unsigned long long)g,
        (as3_v4i_ptr)(unsigned int)(unsigned long long)l,
        0, 0);
#else
    *(v4f*)l = *(const v4f*)g;
#endif
}

static __device__ __forceinline__ void async_join() {
#ifdef HAVE_ASYNC_LDS
    __builtin_amdgcn_s_wait_asynccnt(0);
#endif
}

// ---------------------------------------------------------------------------
// Kernel 1: fused Q/K/V projection.  y = x @ W^T + b  (Linear convention,
// W stored [out,in] row-major so B(k,n) = W[n*K + k] is contiguous in k).
// Macro tile 128(M) x 256(N) per 256-thread block; 8 waves of 64x64 tiles.
// K blocked by 16 through double-buffered LDS (padded stride 20 ->
// conflict-free + aligned); next K-block streams in asynchronously while the
// current one feeds the WMMAs.  Output written directly as [B, H, S, Dh];
// 1/sqrt(Dh) folded into Q.
// ---------------------------------------------------------------------------
#define KB       16
#define LDS_STRD 20   // 16 payload + 4 pad floats per row

__global__ __launch_bounds__(256) void mha_qkv_proj_kernel(
    const float* __restrict__ x,
    const float* __restrict__ wq, const float* __restrict__ bq,
    const float* __restrict__ wk, const float* __restrict__ bk,
    const float* __restrict__ wv, const float* __restrict__ bv,
    float* __restrict__ qo, float* __restrict__ ko, float* __restrict__ vo)
{
    __shared__ float lds_a[2][128 * LDS_STRD];
    __shared__ float lds_b[2][256 * LDS_STRD];

    const int which = blockIdx.z;
    const float* w    = (which == 0) ? wq : (which == 1) ? wk : wv;
    const float* bias = (which == 0) ? bq : (which == 1) ? bk : bv;
    float*       outp = (which == 0) ? qo : (which == 1) ? ko : vo;
    const float  oscale = (which == 0) ? 0.125f : 1.0f;  // 1/sqrt(64) on Q

    const int tid  = threadIdx.x;
    const int wave = tid >> 5;
    const int lane = tid & 31;
    const int half = lane >> 4;
    const int lr   = lane & 15;

    const int blockM = blockIdx.x * 128;
    const int blockN = blockIdx.y * 256;
    const int subM   = (wave >> 2) * 64;   // 0..64
    const int subN   = (wave & 3) * 64;    // 0..192

    // Stage one K-block (A: 128x16, B: 256x16) into LDS buffer `buf`.
    auto stage = [&](int buf, int kb) {
        #pragma unroll
        for (int i = 0; i < 2; ++i) {
            int idx = tid + i * 256;
            int row = idx >> 2;
            int kq  = (idx & 3) * 4;
            copy16(x + (blockM + row) * D_MODEL + kb + kq,
                   &lds_a[buf][row * LDS_STRD + kq]);
        }
        #pragma unroll
        for (int i = 0; i < 4; ++i) {
            int idx = tid + i * 256;
            int n   = idx >> 2;
            int kq  = (idx & 3) * 4;
            copy16(w + (size_t)(blockN + n) * D_MODEL + kb + kq,
                   &lds_b[buf][n * LDS_STRD + kq]);
        }
    };

    v8f acc[4][4];
    for (int mi = 0; mi < 4; ++mi)
        for (int ni = 0; ni < 4; ++ni)
            acc[mi][ni] = (v8f){};

    stage(0, 0);
    async_join();
    __syncthreads();

    for (int kb = 0; kb < D_MODEL; kb += KB) {
        const int cur = (kb >> 4) & 1;
        if (kb + KB < D_MODEL) stage(cur ^ 1, kb + KB);  // overlap with compute

        #pragma unroll
        for (int ks = 0; ks < KB; ks += 4) {
            const int kk = ks + half * 2;  // even -> 8B aligned LDS reads
            v2f afrag[4], bfrag[4];
            #pragma unroll
            for (int mi = 0; mi < 4; ++mi)
                afrag[mi] = *(const v2f*)(&lds_a[cur][(subM + mi * 16 + lr) * LDS_STRD + kk]);
            #pragma unroll
            for (int ni = 0; ni < 4; ++ni)
                bfrag[ni] = *(const v2f*)(&lds_b[cur][(subN + ni * 16 + lr) * LDS_STRD + kk]);
            #pragma unroll
            for (int mi = 0; mi < 4; ++mi)
                #pragma unroll
                for (int ni = 0; ni < 4; ++ni)
                    acc[mi][ni] = wmma_f32(afrag[mi], bfrag[ni], acc[mi][ni]);
        }

        async_join();      // own async copies into the other buffer done
        __syncthreads();   // everyone done copying + done reading `cur`
    }

    // Epilogue: bias, scale, scatter into [B, H, S, Dh].
    #pragma unroll
    for (int ni = 0; ni < 4; ++ni) {
        const int n  = blockN + subN + ni * 16 + lr;
        const float bn = bias[n];
        const int h  = n >> 6;
        const int dh = n & 63;
        #pragma unroll
        for (int mi = 0; mi < 4; ++mi) {
            #pragma unroll
            for (int r = 0; r < 8; ++r) {
                const int m = blockM + subM + mi * 16 + half * 8 + r;
                const int b = m >> 11;       // / SEQ
                const int s = m & (SEQ - 1);
                const float val = (acc[mi][ni][r] + bn) * oscale;
                outp[((size_t)(b * N_HEADS + h) * SEQ + s) * D_HEAD + dh] = val;
            }
        }
    }
}

// ---------------------------------------------------------------------------
// Kernel 2: causal flash attention.  One wave per (b, h, 16-row query tile).
// S = Q Kt via 16 f32 WMMAs (K = Dh = 64); online softmax with 16-lane
// shfl_xor row reductions; P converted C-layout -> A-layout via a wave-
// private LDS tile; O += P V via 16 WMMAs into 4 accumulator fragments.
// ---------------------------------------------------------------------------
__global__ __launch_bounds__(256) void mha_attention_kernel(
    const float* __restrict__ q, const float* __restrict__ k,
    const float* __restrict__ v, const unsigned char* __restrict__ mask,
    float* __restrict__ ctx)
{
    __shared__ float lds_p[8][16 * 18];  // stride 18: conflict-free, 8B aligned

    const int tid  = threadIdx.x;
    const int wave = tid >> 5;
    const int lane = tid & 31;
    const int half = lane >> 4;
    const int lr   = lane & 15;

    const int wid = blockIdx.x * 8 + wave;        // 0..4095
    const int bh  = wid >> 7;                     // 0..31  (b*H + h)
    const int qt  = wid & 127;                    // query tile 0..127
    const int b   = bh >> 4;
    const int q0  = qt * 16;
    const size_t base = (size_t)bh * SEQ * D_HEAD;

    // Preload Q A-fragments (Q already scaled by 1/sqrt(Dh)).
    v2f aq[16];
    #pragma unroll
    for (int ks = 0; ks < 16; ++ks)
        aq[ks] = *(const v2f*)(q + base + (size_t)(q0 + lr) * D_HEAD
                               + ks * 4 + half * 2);

    v8f o[4];
    #pragma unroll
    for (int t = 0; t < 4; ++t) o[t] = (v8f){};
    float mrun[8], lrun[8];
    #pragma unroll
    for (int r = 0; r < 8; ++r) { mrun[r] = -__builtin_inff(); lrun[r] = 0.0f; }

    for (int kt = 0; kt <= qt; ++kt) {
        const int k0 = kt * 16;

        // Prefetch next iteration's K/V rows (global_prefetch_b8).
        if (kt < qt) {
            __builtin_prefetch(k + base + (size_t)(k0 + 16 + lr) * D_HEAD, 0, 0);
            __builtin_prefetch(v + base + (size_t)(k0 + 16 + lr) * D_HEAD, 0, 0);
        }

        // ---- scores S = Q @ K^T  (B(d,n) = K[k0+n][d], contiguous in d) ----
        v8f s = (v8f){};
        #pragma unroll
        for (int ks = 0; ks < 16; ++ks) {
            v2f bk = *(const v2f*)(k + base + (size_t)(k0 + lr) * D_HEAD
                                   + ks * 4 + half * 2);
            s = wmma_f32(aq[ks], bk, s);
        }

        // ---- masking + online softmax (per row m = half*8 + r) ----
        const bool colmask = mask[(size_t)b * SEQ + k0 + lr] != 0;
        const bool diag    = (kt == qt);
        #pragma unroll
        for (int r = 0; r < 8; ++r) {
            const int mrow = half * 8 + r;
            float sv = s[r];
            if (colmask || (diag && (lr > mrow))) sv = -__builtin_inff();
            float rm = sv;
            #pragma unroll
            for (int off = 8; off >= 1; off >>= 1)
                rm = fmaxf(rm, __shfl_xor(rm, off, 32));
            const float mnew  = fmaxf(mrun[r], rm);
            const float alpha = __expf(mrun[r] - mnew);   // exp(-inf)=0 on 1st hit
            const float p     = __expf(sv - mnew);
            float rs = p;
            #pragma unroll
            for (int off = 8; off >= 1; off >>= 1)
                rs += __shfl_xor(rs, off, 32);
            lrun[r] = lrun[r] * alpha + rs;
            mrun[r] = mnew;
            #pragma unroll
            for (int t = 0; t < 4; ++t) o[t][r] *= alpha;
            s[r] = p;
        }

        // ---- C-layout P -> A-layout via wave-private LDS tile ----
        float* pl = &lds_p[wave][0];
        #pragma unroll
        for (int r = 0; r < 8; ++r)
            pl[(half * 8 + r) * 18 + lr] = s[r];
        // wave-private LDS: per-wave DS ordering + compiler dscnt waits suffice
        v2f ap[4];
        #pragma unroll
        for (int ks = 0; ks < 4; ++ks)
            ap[ks] = *(const v2f*)(&pl[lr * 18 + ks * 4 + half * 2]);

        // ---- O += P @ V ----
        #pragma unroll
        for (int t = 0; t < 4; ++t) {
            #pragma unroll
            for (int ks = 0; ks < 4; ++ks) {
                const int kk = ks * 4 + half * 2;
                v2f bv;
                bv[0] = v[base + (size_t)(k0 + kk)     * D_HEAD + t * 16 + lr];
                bv[1] = v[base + (size_t)(k0 + kk + 1) * D_HEAD + t * 16 + lr];
                o[t] = wmma_f32(ap[ks], bv, o[t]);
            }
        }
    }

    // ---- normalize and store context [B, H, S, Dh] ----
    #pragma unroll
    for (int r = 0; r < 8; ++r) {
        const float inv = 1.0f / lrun[r];
        #pragma unroll
        for (int t = 0; t < 4; ++t)
            ctx[base + (size_t)(q0 + half * 8 + r) * D_HEAD + t * 16 + lr]
                = o[t][r] * inv;
    }
}

// ---------------------------------------------------------------------------
// Kernel 3: output projection.  Same double-buffered async GEMM core; A rows
// gathered from the [B, H, S, Dh] context layout (16-wide k-runs stay inside
// one head since KB=16 divides Dh=64, so b128 copies remain contiguous).
// ---------------------------------------------------------------------------
__global__ __launch_bounds__(256) void mha_out_proj_kernel(
    const float* __restrict__ ctx, const float* __restrict__ wo,
    const float* __restrict__ bo, float* __restrict__ out)
{
    __shared__ float lds_a[2][128 * LDS_STRD];
    __shared__ float lds_b[2][256 * LDS_STRD];

    const int tid  = threadIdx.x;
    const int wave = tid >> 5;
    const int lane = tid & 31;
    const int half = lane >> 4;
    const int lr   = lane & 15;

    const int blockM = blockIdx.x * 128;
    const int blockN = blockIdx.y * 256;
    const int subM   = (wave >> 2) * 64;
    const int subN   = (wave & 3) * 64;

    auto stage = [&](int buf, int kb) {
        const int h = kb >> 6;  // head constant across a 16-wide k-run
        #pragma unroll
        for (int i = 0; i < 2; ++i) {
            int idx = tid + i * 256;
            int row = idx >> 2;
            int kq  = (idx & 3) * 4;
            int m   = blockM + row;
            int bb  = m >> 11;
            int ss  = m & (SEQ - 1);
            int dh  = (kb & 63) + kq;
            copy16(ctx + ((size_t)(bb * N_HEADS + h) * SEQ + ss) * D_HEAD + dh,
                   &lds_a[buf][row * LDS_STRD + kq]);
        }
        #pragma unroll
        for (int i = 0; i < 4; ++i) {
            int idx = tid + i * 256;
            int n   = idx >> 2;
            int kq  = (idx & 3) * 4;
            copy16(wo + (size_t)(blockN + n) * D_MODEL + kb + kq,
                   &lds_b[buf][n * LDS_STRD + kq]);
        }
    };

    v8f acc[4][4];
    for (int mi = 0; mi < 4; ++mi)
        for (int ni = 0; ni < 4; ++ni)
            acc[mi][ni] = (v8f){};

    stage(0, 0);
    async_join();
    __syncthreads();

    for (int kb = 0; kb < D_MODEL; kb += KB) {
        const int cur = (kb >> 4) & 1;
        if (kb + KB < D_MODEL) stage(cur ^ 1, kb + KB);

        #pragma unroll
        for (int ks = 0; ks < KB; ks += 4) {
            const int kk = ks + half * 2;
            v2f afrag[4], bfrag[4];
            #pragma unroll
            for (int mi = 0; mi < 4; ++mi)
                afrag[mi] = *(const v2f*)(&lds_a[cur][(subM + mi * 16 + lr) * LDS_STRD + kk]);
            #pragma unroll
            for (int ni = 0; ni < 4; ++ni)
                bfrag[ni] = *(const v2f*)(&lds_b[cur][(subN + ni * 16 + lr) * LDS_STRD + kk]);
            #pragma unroll
            for (int mi = 0; mi < 4; ++mi)
                #pragma unroll
                for (int ni = 0; ni < 4; ++ni)
                    acc[mi][ni] = wmma_f32(afrag[mi], bfrag[ni], acc[mi][ni]);
        }

        async_join();
        __syncthreads();
    }

    #pragma unroll
    for (int ni = 0; ni < 4; ++ni) {
        const int n  = blockN + subN + ni * 16 + lr;
        const float bn = bo[n];
        #pragma unroll
        for (int mi = 0; mi < 4; ++mi) {
            #pragma unroll
            for (int r = 0; r < 8; ++r) {
                const int m = blockM + subM + mi * 16 + half * 8 + r;
                out[(size_t)m * D_MODEL + n] = acc[mi][ni][r] + bn;
            }
        }
    }
}

// ---------------------------------------------------------------------------
// Host-side launcher.
// Inputs (setup_inputs order): x, mask, wq, bq, wk, bk, wv, bv, wo, bo.
// Workspace: Q | K | V | CTX, each B*H*S*Dh = 4,194,304 floats (16 MB).
// ---------------------------------------------------------------------------
extern "C" void kernel_launch(void* const* d_in, const int* in_sizes, int n_in,
                              void* d_out, int out_size, void* d_ws, size_t ws_size,
                              hipStream_t stream) {
    const float*         x    = (const float*)d_in[0];
    const unsigned char* mask = (const unsigned char*)d_in[1];  // bool array
    const float* wq = (const float*)d_in[2];
    const float* bq = (const float*)d_in[3];
    const float* wk = (const float*)d_in[4];
    const float* bk = (const float*)d_in[5];
    const float* wv = (const float*)d_in[6];
    const float* bv = (const float*)d_in[7];
    const float* wo = (const float*)d_in[8];
    const float* bo = (const float*)d_in[9];
    float* out = (float*)d_out;

    const size_t per = (size_t)BATCH * N_HEADS * SEQ * D_HEAD;  // 4,194,304
    float* qw = (float*)d_ws;
    float* kw = qw + per;
    float* vw = kw + per;
    float* cw = vw + per;

    // QKV projections: M=4096, N=1024, macro tile 128x256, z selects Q/K/V.
    mha_qkv_proj_kernel<<<dim3(M_TOTAL / 128, D_MODEL / 256, 3), 256, 0, stream>>>(
        x, wq, bq, wk, bk, wv, bv, qw, kw, vw);

    // Flash attention: 32 (b,h) * 128 q-tiles = 4096 waves / 8 per block.
    mha_attention_kernel<<<dim3(512), 256, 0, stream>>>(qw, kw, vw, mask, cw);

    // Output projection.
    mha_out_proj_kernel<<<dim3(M_TOTAL / 128, D_MODEL / 256), 256, 0, stream>>>(
        cw, wo, bo, out);
}